// MambaTextClassification_71777493450827
// MI455X (gfx1250) — compile-verified
//
#include <hip/hip_runtime.h>
#include <cmath>

#define Dm   768
#define NLm  8
#define DIm  1536
#define DSm  16
#define DCm  4
#define DTRm 48
#define DBCW 80          // DTR + 2*DS
#define NCm  4
#define Bm   2
#define Lm   1024
#define MROWS (Bm*Lm)    // 2048

typedef float v2f __attribute__((ext_vector_type(2)));
typedef float v8f __attribute__((ext_vector_type(8)));

// ---------------- embedding gather ----------------
__global__ void embed_kernel(const int* __restrict__ ids,
                             const float* __restrict__ embed,
                             float* __restrict__ x, int total) {
    int idx = blockIdx.x * blockDim.x + threadIdx.x;
    if (idx >= total) return;
    int row = idx / Dm;          // b*L + l
    int d   = idx - row * Dm;
    x[idx] = embed[(long)ids[row] * Dm + d];
}

// ---------------- rmsnorm (one block per row of 768) ----------------
__global__ void rmsnorm_kernel(const float* __restrict__ x,
                               const float* __restrict__ w,
                               float* __restrict__ out) {
    __shared__ float red[8];
    int row = blockIdx.x;
    const float* xr = x + (long)row * Dm;
    float s = 0.f;
    for (int d = threadIdx.x; d < Dm; d += 256) { float v = xr[d]; s += v * v; }
    for (int off = 16; off > 0; off >>= 1) s += __shfl_down(s, off);
    if ((threadIdx.x & 31) == 0) red[threadIdx.x >> 5] = s;
    __syncthreads();
    if (threadIdx.x == 0) {
        float t = 0.f;
        for (int i = 0; i < 8; ++i) t += red[i];
        red[0] = rsqrtf(t / (float)Dm + 1e-5f);
    }
    __syncthreads();
    float r = red[0];
    for (int d = threadIdx.x; d < Dm; d += 256)
        out[(long)row * Dm + d] = xr[d] * r * w[d];
}

// ---------------- WMMA f32 GEMM:  D = A(MxK, lda) * W(NxK)^T ----------------
// Wave tile: (16*MSUB) x (16*NSUB); software-pipelined fragment loads.
// EPI 0: plain store   EPI 1: softplus(acc + bias[n])   EPI 2: acc + resid[m*ldd+n]
template <int MSUB, int NSUB, int EPI>
__global__ void gemm_wmma(const float* __restrict__ A, int lda,
                          const float* __restrict__ W,
                          float* __restrict__ Dst,
                          const float* __restrict__ bias,
                          const float* __restrict__ resid,
                          int M, int N, int K, int ldd) {
    int wave = (blockIdx.x * blockDim.x + threadIdx.x) >> 5;
    int lane = threadIdx.x & 31;
    int tilesN = N / (16 * NSUB);
    int mt = wave / tilesN;
    int nt = wave - mt * tilesN;
    if (mt >= M / (16 * MSUB)) return;
    int m0 = mt * 16 * MSUB;
    int n0 = nt * 16 * NSUB;

    int ml = lane & 15;              // A: row M; W: row N; D: col N
    int kh = (lane >> 4) << 1;       // 0 or 2  (K sub-offset per lane half)

    v8f acc[MSUB][NSUB] = {};
    const float* arow[MSUB];
    const float* wrow[NSUB];
    #pragma unroll
    for (int i = 0; i < MSUB; ++i)
        arow[i] = A + (long)(m0 + i * 16 + ml) * lda + kh;
    #pragma unroll
    for (int j = 0; j < NSUB; ++j)
        wrow[j] = W + (long)(n0 + j * 16 + ml) * K + kh;

    // prologue: fragment buffers for k = 0
    v2f a0[MSUB], b0[NSUB];
    #pragma unroll
    for (int i = 0; i < MSUB; ++i) a0[i] = *(const v2f*)(arow[i]);
    #pragma unroll
    for (int j = 0; j < NSUB; ++j) b0[j] = *(const v2f*)(wrow[j]);

    for (int k = 0; k < K; k += 4) {
        // prefetch next K-step into separate registers (clamped: last iter reloads k=0)
        int kp = (k + 4 < K) ? (k + 4) : 0;
        v2f a1[MSUB], b1[NSUB];
        #pragma unroll
        for (int i = 0; i < MSUB; ++i) a1[i] = *(const v2f*)(arow[i] + kp);
        #pragma unroll
        for (int j = 0; j < NSUB; ++j) b1[j] = *(const v2f*)(wrow[j] + kp);

        // MSUB*NSUB independent accumulation chains
        #pragma unroll
        for (int i = 0; i < MSUB; ++i)
            #pragma unroll
            for (int j = 0; j < NSUB; ++j)
                acc[i][j] = __builtin_amdgcn_wmma_f32_16x16x4_f32(
                    false, a0[i], false, b0[j], (short)0, acc[i][j], false, false);

        #pragma unroll
        for (int i = 0; i < MSUB; ++i) a0[i] = a1[i];
        #pragma unroll
        for (int j = 0; j < NSUB; ++j) b0[j] = b1[j];
    }

    int mh = (lane >> 4) << 3;       // 0 or 8 (M sub-offset for C/D)
    #pragma unroll
    for (int i = 0; i < MSUB; ++i) {
        #pragma unroll
        for (int j = 0; j < NSUB; ++j) {
            int n = n0 + j * 16 + ml;
            #pragma unroll
            for (int r = 0; r < 8; ++r) {
                int m = m0 + i * 16 + mh + r;
                float v = acc[i][j][r];
                if (EPI == 1) {
                    float t = v + bias[n];
                    v = fmaxf(t, 0.f) + log1pf(__expf(-fabsf(t)));   // stable softplus
                } else if (EPI == 2) {
                    v += resid[(long)m * ldd + n];
                }
                Dst[(long)m * ldd + n] = v;
            }
        }
    }
}

// ---------------- depthwise causal conv (DC=4) + SiLU ----------------
__global__ void conv_silu_kernel(const float* __restrict__ xz,   // (M, 2*DI), x part at [0:DI)
                                 const float* __restrict__ cw,   // (DI, DC)
                                 const float* __restrict__ cb,   // (DI)
                                 float* __restrict__ xc, int total) {
    int idx = blockIdx.x * blockDim.x + threadIdx.x;
    if (idx >= total) return;                 // total = M * DI
    int i   = idx % DIm;
    int row = idx / DIm;                      // b*L + l
    int l   = row % Lm;
    float acc = cb[i];
    #pragma unroll
    for (int k = 0; k < DCm; ++k) {
        int ls = l + k - (DCm - 1);
        if (ls >= 0)
            acc += xz[(long)(row + ls - l) * (2 * DIm) + i] * cw[i * DCm + k];
    }
    float sig = 1.f / (1.f + __expf(-acc));
    xc[(long)row * DIm + i] = acc * sig;
}

// ---------------- selective scan: one thread per (b, channel) ----------------
__global__ void scan_kernel(const float* __restrict__ dbc,   // (M, 80): [48:64)=B, [64:80)=C
                            const float* __restrict__ dt,    // (M, DI)
                            const float* __restrict__ xc,    // (M, DI)
                            const float* __restrict__ xz,    // (M, 2*DI), z at [DI:2DI)
                            const float* __restrict__ A_log, // (DI, DS) for this layer
                            const float* __restrict__ Dskip, // (DI)
                            float* __restrict__ y) {         // (M, DI)
    int idx = blockIdx.x * blockDim.x + threadIdx.x;
    if (idx >= Bm * DIm) return;
    int b = idx / DIm, i = idx % DIm;
    float As[DSm];
    #pragma unroll
    for (int s = 0; s < DSm; ++s) As[s] = -__expf(A_log[i * DSm + s]);
    float Dk = Dskip[i];
    float h[DSm];
    #pragma unroll
    for (int s = 0; s < DSm; ++s) h[s] = 0.f;

    for (int l = 0; l < Lm; ++l) {
        int row = b * Lm + l;
        float dtv = dt[(long)row * DIm + i];
        float xv  = xc[(long)row * DIm + i];
        const float* Bt = dbc + (long)row * DBCW + DTRm;
        const float* Ct = Bt + DSm;
        float dx = dtv * xv;
        float yv = 0.f;
        #pragma unroll
        for (int s = 0; s < DSm; ++s) {
            h[s] = h[s] * __expf(dtv * As[s]) + dx * Bt[s];
            yv  += h[s] * Ct[s];
        }
        float zv = xz[(long)row * (2 * DIm) + DIm + i];
        float silu_z = zv / (1.f + __expf(-zv));
        y[(long)row * DIm + i] = (yv + xv * Dk) * silu_z;
    }
}

// ---------------- mean over L ----------------
__global__ void mean_kernel(const float* __restrict__ xn, float* __restrict__ mean) {
    int idx = blockIdx.x * blockDim.x + threadIdx.x;     // B*D threads
    if (idx >= Bm * Dm) return;
    int b = idx / Dm, d = idx % Dm;
    float s = 0.f;
    for (int l = 0; l < Lm; ++l) s += xn[(long)(b * Lm + l) * Dm + d];
    mean[idx] = s * (1.f / (float)Lm);
}

// ---------------- classifier head ----------------
__global__ void logits_kernel(const float* __restrict__ mean,
                              const float* __restrict__ cw,
                              const float* __restrict__ cb,
                              float* __restrict__ out) {
    int tid = threadIdx.x;
    if (tid >= Bm * NCm) return;
    int b = tid / NCm, c = tid % NCm;
    float s = cb[c];
    for (int d = 0; d < Dm; ++d) s += mean[b * Dm + d] * cw[c * Dm + d];
    out[tid] = s;
}

extern "C" void kernel_launch(void* const* d_in, const int* in_sizes, int n_in,
                              void* d_out, int out_size, void* d_ws, size_t ws_size,
                              hipStream_t stream) {
    const int*   ids      = (const int*)  d_in[0];
    const float* embed    = (const float*)d_in[1];
    const float* norm_w   = (const float*)d_in[2];
    const float* in_w     = (const float*)d_in[3];
    const float* conv_w   = (const float*)d_in[4];
    const float* conv_b   = (const float*)d_in[5];
    const float* xp_w     = (const float*)d_in[6];
    const float* dt_w     = (const float*)d_in[7];
    const float* dt_b     = (const float*)d_in[8];
    const float* A_log    = (const float*)d_in[9];
    const float* D_skip   = (const float*)d_in[10];
    const float* out_w    = (const float*)d_in[11];
    const float* norm_f_w = (const float*)d_in[12];
    const float* cls_w    = (const float*)d_in[13];
    const float* cls_b    = (const float*)d_in[14];

    // workspace layout
    float* x    = (float*)d_ws;                 // (M, D)
    float* xn   = x    + (long)MROWS * Dm;      // (M, D)
    float* xz   = xn   + (long)MROWS * Dm;      // (M, 2*DI)
    float* xc   = xz   + (long)MROWS * 2 * DIm; // (M, DI)
    float* dbc  = xc   + (long)MROWS * DIm;     // (M, 80)
    float* dtb  = dbc  + (long)MROWS * DBCW;    // (M, DI)
    float* yb   = dtb  + (long)MROWS * DIm;     // (M, DI)
    float* mnb  = yb   + (long)MROWS * DIm;     // (B, D)

    auto gblk = [](long waves) { return (int)((waves * 32 + 255) / 256); };

    // x = embed[ids]
    {
        int total = MROWS * Dm;
        embed_kernel<<<(total + 255) / 256, 256, 0, stream>>>(ids, embed, x, total);
    }

    for (int l = 0; l < NLm; ++l) {
        const float* nw  = norm_w + (long)l * Dm;
        const float* iw  = in_w   + (long)l * 2 * DIm * Dm;
        const float* cwl = conv_w + (long)l * DIm * DCm;
        const float* cbl = conv_b + (long)l * DIm;
        const float* xpl = xp_w   + (long)l * DBCW * DIm;
        const float* dwl = dt_w   + (long)l * DIm * DTRm;
        const float* dbl = dt_b   + (long)l * DIm;
        const float* al  = A_log  + (long)l * DIm * DSm;
        const float* dsl = D_skip + (long)l * DIm;
        const float* owl = out_w  + (long)l * Dm * DIm;

        // xn = rmsnorm(x)
        rmsnorm_kernel<<<MROWS, 256, 0, stream>>>(x, nw, xn);

        // xz = xn @ in_w^T   (M=2048, N=3072, K=768); 32x64 wave tiles
        gemm_wmma<2, 4, 0><<<gblk((MROWS / 32) * (2 * DIm / 64)), 256, 0, stream>>>(
            xn, Dm, iw, xz, nullptr, nullptr, MROWS, 2 * DIm, Dm, 2 * DIm);

        // xc = silu(causal_conv(xz[:, :DI]) + conv_b)
        {
            int total = MROWS * DIm;
            conv_silu_kernel<<<(total + 255) / 256, 256, 0, stream>>>(xz, cwl, cbl, xc, total);
        }

        // dbc = xc @ xp_w^T   (N=80, K=1536); 32x16 wave tiles
        gemm_wmma<2, 1, 0><<<gblk((MROWS / 32) * (DBCW / 16)), 256, 0, stream>>>(
            xc, DIm, xpl, dbc, nullptr, nullptr, MROWS, DBCW, DIm, DBCW);

        // dt = softplus(dbc[:, :48] @ dt_w^T + dt_b)   (N=1536, K=48, lda=80)
        gemm_wmma<2, 4, 1><<<gblk((MROWS / 32) * (DIm / 64)), 256, 0, stream>>>(
            dbc, DBCW, dwl, dtb, dbl, nullptr, MROWS, DIm, DTRm, DIm);

        // selective scan + D-skip + silu(z) gating -> yb
        scan_kernel<<<(Bm * DIm + 255) / 256, 256, 0, stream>>>(
            dbc, dtb, xc, xz, al, dsl, yb);

        // x = x + yb @ out_w^T   (N=768, K=1536, fused residual)
        gemm_wmma<2, 4, 2><<<gblk((MROWS / 32) * (Dm / 64)), 256, 0, stream>>>(
            yb, DIm, owl, x, nullptr, x, MROWS, Dm, DIm, Dm);
    }

    // final rmsnorm, mean over L, classifier
    rmsnorm_kernel<<<MROWS, 256, 0, stream>>>(x, norm_f_w, xn);
    mean_kernel<<<(Bm * Dm + 255) / 256, 256, 0, stream>>>(xn, mnb);
    logits_kernel<<<1, 64, 0, stream>>>(mnb, cls_w, cls_b, (float*)d_out);
}